// BilinearSeqAttnAction3_11175504904501
// MI455X (gfx1250) — compile-verified
//
#include <hip/hip_runtime.h>

// Problem constants (match the reference).
#define BB 8
#define LL 2048
#define HH 2048
#define KT_SPLIT 16          // stage-1 split-K chunks (H/128)
#define NEG_INF_F (-1.0e30f)

typedef __attribute__((ext_vector_type(2))) float v2f;
typedef __attribute__((ext_vector_type(4))) float f32x4;
typedef __attribute__((ext_vector_type(8))) float v8f;

__device__ __forceinline__ float fsig(float v) {
    // sigmoid via fast HW transcendentals: 1 / (1 + 2^(-x*log2e))
    return __builtin_amdgcn_rcpf(1.0f + __builtin_amdgcn_exp2f(-1.44269504f * v));
}

// ---------------------------------------------------------------------------
// Kernel A: stage-1 split-K partials.
//   partial[b, kt, j] = sum_{h in chunk kt} y[b,h] * weight[h,j] * sigmoid(wa[a_b,h,j])
// Grid (2 j-tiles, 16 k-chunks, B) x 256 threads; each thread owns 4 j's (b128 loads).
// wa is single-touch -> non-temporal loads so the 16MB `weight` (reused by all
// 8 batches) stays hot in the 192MB L2.
// ---------------------------------------------------------------------------
__global__ void __launch_bounds__(256)
wy_partial_k(const float* __restrict__ y, const int* __restrict__ actions,
             const float* __restrict__ weight, const float* __restrict__ wa,
             float* __restrict__ partial) {
    const int tid = threadIdx.x;
    const int jt = blockIdx.x, kt = blockIdx.y, b = blockIdx.z;
    const int a = actions[b];
    const int j  = jt * 1024 + tid * 4;
    const int h0 = kt * (HH / KT_SPLIT);

    __shared__ float ys[HH / KT_SPLIT];
    if (tid < (HH / KT_SPLIT)) ys[tid] = y[(size_t)b * HH + h0 + tid];
    __syncthreads();

    const float* wptr  = weight + (size_t)h0 * HH + j;
    const float* waptr = wa + ((size_t)a * HH + h0) * HH + j;

    f32x4 acc = {0.f, 0.f, 0.f, 0.f};
#pragma unroll 4
    for (int h = 0; h < HH / KT_SPLIT; ++h) {
        f32x4 w = *(const f32x4*)wptr;
        f32x4 g = __builtin_nontemporal_load((const f32x4*)waptr);   // single-touch stream
        f32x4 s;
        s.x = fsig(g.x); s.y = fsig(g.y); s.z = fsig(g.z); s.w = fsig(g.w);
        acc += ys[h] * w * s;
        wptr  += HH;
        waptr += HH;
    }
    *(f32x4*)(partial + ((size_t)(b * KT_SPLIT + kt)) * HH + j) = acc;
}

// ---------------------------------------------------------------------------
// Kernel A2: reduce split-K partials + gated bias -> Wy[b,j]
// ---------------------------------------------------------------------------
__global__ void __launch_bounds__(256)
wy_reduce_k(const float* __restrict__ partial, const int* __restrict__ actions,
            const float* __restrict__ bias, const float* __restrict__ ba,
            float* __restrict__ Wy) {
    const int idx = blockIdx.x * 256 + threadIdx.x;      // 0 .. B*H-1
    const int b = idx >> 11, j = idx & (HH - 1);
    float s = 0.f;
#pragma unroll
    for (int t = 0; t < KT_SPLIT; ++t)
        s += partial[((size_t)(b * KT_SPLIT + t)) * HH + j];
    const int a = actions[b];
    s += bias[j] * fsig(ba[(size_t)a * HH + j]);
    Wy[idx] = s;
}

// ---------------------------------------------------------------------------
// Kernel B: stage-2 contraction on the matrix pipe, full f32 precision.
//   xWy[b,l] = x[b,l,:] . Wy[b,:]    via V_WMMA_F32_16X16X4_F32
// One wave owns a 16-row tile; 512 WMMAs accumulate K=2048.
//   A (16x4 f32):  lanes 0-15 hold VGPR0=K0,VGPR1=K1; lanes 16-31 hold K2,K3
//                  -> lane loads x[row=lane%16, k0+2*(lane/16) .. +1] as one b64.
//                  Consecutive k0 reuse the same 128B lines in WGP$ (8 iters/line).
//   B (4x16 f32):  Wy K-chunk broadcast across all 16 N columns (LDS broadcast,
//                  conflict-free) with the matching per-lane K pairing.
// Every D column then equals the dot product; lanes 0/16 write the 16 rows.
// Bandwidth-bound => the 15 redundant columns cost nothing, VALU stays free.
// ---------------------------------------------------------------------------
__global__ void __launch_bounds__(128)
xwy_wmma_k(const float* __restrict__ x, const float* __restrict__ Wy,
           float* __restrict__ xwy) {
    const int b = blockIdx.y;
    const int tid = threadIdx.x;
    const int wave = tid >> 5, lane = tid & 31;
    const int m = lane & 15, half = lane >> 4;

    __shared__ float wys[HH];
    for (int i = tid; i < HH; i += 128) wys[i] = Wy[(size_t)b * HH + i];
    __syncthreads();

    const int l0 = (blockIdx.x * 4 + wave) * 16;
    const float* xrow = x + ((size_t)b * LL + l0 + m) * HH + 2 * half;

    v8f c = {};
#pragma unroll 8
    for (int k0 = 0; k0 < HH; k0 += 4) {
        v2f a = *(const v2f*)(xrow + k0);                 // A: 16 rows x K=4 tile
        v2f bm = *(const v2f*)(&wys[k0 + 2 * half]);      // B: Wy chunk, bcast over N
        c = __builtin_amdgcn_wmma_f32_16x16x4_f32(
                /*neg_a=*/false, a, /*neg_b=*/false, bm,
                /*c_mod=*/(short)0, c, /*reuse_a=*/false, /*reuse_b=*/false);
    }
    // C/D layout: lanes 0-15 hold M=i in c[i]; lanes 16-31 hold M=i+8.
    if (lane == 0 || lane == 16) {
        float* o = xwy + (size_t)b * LL + l0 + half * 8;
#pragma unroll
        for (int i = 0; i < 8; ++i) o[i] = c[i];
    }
}

// ---------------------------------------------------------------------------
// Kernel C: masked softmax over L per batch row. One 256-thread block per b.
// ---------------------------------------------------------------------------
__global__ void __launch_bounds__(256)
softmax_k(const float* __restrict__ xwy, const unsigned char* __restrict__ mask,
          float* __restrict__ out) {
    const int b = blockIdx.x, tid = threadIdx.x;
    __shared__ float red[256];
    const size_t base = (size_t)b * LL;

    float v[LL / 256];
    float mx = -3.0e38f;
#pragma unroll
    for (int i = 0; i < LL / 256; ++i) {
        const int idx = tid + i * 256;
        float t = xwy[base + idx];
        t = mask[base + idx] ? NEG_INF_F : t;
        v[i] = t;
        mx = fmaxf(mx, t);
    }
    red[tid] = mx;
    __syncthreads();
    for (int s = 128; s > 0; s >>= 1) {
        if (tid < s) red[tid] = fmaxf(red[tid], red[tid + s]);
        __syncthreads();
    }
    mx = red[0];
    __syncthreads();

    float sum = 0.f;
#pragma unroll
    for (int i = 0; i < LL / 256; ++i) {
        float e = __builtin_amdgcn_exp2f((v[i] - mx) * 1.44269504f);
        v[i] = e;
        sum += e;
    }
    red[tid] = sum;
    __syncthreads();
    for (int s = 128; s > 0; s >>= 1) {
        if (tid < s) red[tid] += red[tid + s];
        __syncthreads();
    }
    const float inv = __builtin_amdgcn_rcpf(red[0]);
#pragma unroll
    for (int i = 0; i < LL / 256; ++i)
        out[base + tid + i * 256] = v[i] * inv;
}

// ---------------------------------------------------------------------------
extern "C" void kernel_launch(void* const* d_in, const int* in_sizes, int n_in,
                              void* d_out, int out_size, void* d_ws, size_t ws_size,
                              hipStream_t stream) {
    const float*         x       = (const float*)d_in[0];          // [B,L,H]
    const float*         y       = (const float*)d_in[1];          // [B,H]
    const unsigned char* x_mask  = (const unsigned char*)d_in[2];  // [B,L] bool
    const int*           actions = (const int*)d_in[3];            // [B]
    const float*         weight  = (const float*)d_in[4];          // [H,H]
    const float*         bias    = (const float*)d_in[5];          // [H]
    const float*         wa      = (const float*)d_in[6];          // [A,H,H]
    const float*         ba      = (const float*)d_in[7];          // [A,H]
    float*               out     = (float*)d_out;                  // [B,L]

    // Workspace layout: partials (1 MB) | Wy (64 KB) | logits (64 KB)
    float* partial = (float*)d_ws;                                 // B*KT_SPLIT*H
    float* Wy      = partial + (size_t)BB * KT_SPLIT * HH;         // B*H
    float* logits  = Wy + (size_t)BB * HH;                         // B*L

    dim3 gA(2, KT_SPLIT, BB);
    wy_partial_k<<<gA, 256, 0, stream>>>(y, actions, weight, wa, partial);

    wy_reduce_k<<<(BB * HH) / 256, 256, 0, stream>>>(partial, actions, bias, ba, Wy);

    dim3 gB(LL / 64, BB);
    xwy_wmma_k<<<gB, 128, 0, stream>>>(x, Wy, logits);

    softmax_k<<<BB, 256, 0, stream>>>(logits, x_mask, out);
}